// Adaptive_temperature_scale_61624190763241
// MI455X (gfx1250) — compile-verified
//
#include <hip/hip_runtime.h>
#include <cmath>

#define NROWS 524288
#define NCOLS 128
#define ROWS_PER_BLOCK 128
#define NBLOCKS (NROWS / ROWS_PER_BLOCK)   // 4096

typedef float v2f __attribute__((ext_vector_type(2)));
typedef float v8f __attribute__((ext_vector_type(8)));

__device__ __forceinline__ float wave_sum32(float v) {
#pragma unroll
  for (int off = 16; off > 0; off >>= 1) v += __shfl_xor(v, off);
  return v;
}

// One wave (32 lanes) handles 16 rows. Lane l works on row (l & 15); the two
// half-waves split the K dimension exactly as the WMMA f32 16x16x4 A-fragment:
//   lane l < 16 : K = 4*kk + {0,1};  lane l >= 16 : K = 4*kk + {2,3}
// B is w_L replicated across the 16 N columns, so D[m][n] = LTS[m] for all n.
__global__ __launch_bounds__(256) void ats_main(
    const float* __restrict__ X, const int* __restrict__ labels,
    const float* __restrict__ wL, const float* __restrict__ wHp,
    const float* __restrict__ bp, float* __restrict__ out,
    float* __restrict__ blockSums)
{
  __shared__ float sred[8];
  const int tid  = threadIdx.x;
  const int wave = tid >> 5;
  const int lane = tid & 31;
  const int m    = lane & 15;
  const int half = lane >> 4;
  const int row  = blockIdx.x * ROWS_PER_BLOCK + wave * 16 + m;
  const size_t rowOff = (size_t)row * NCOLS;
  const int cb = 2 * half;

  // ---- single pass load + WMMA mat-vec (LTS) -------------------------------
  v2f xr[32];
  v8f acc = {0.f, 0.f, 0.f, 0.f, 0.f, 0.f, 0.f, 0.f};
#pragma unroll
  for (int kk = 0; kk < 32; ++kk) {
    v2f xv = *(const v2f*)(X  + rowOff + 4 * kk + cb);
    v2f wv = *(const v2f*)(wL +          4 * kk + cb);
    xr[kk] = xv;
    acc = __builtin_amdgcn_wmma_f32_16x16x4_f32(
        /*neg_a=*/false, xv, /*neg_b=*/false, wv,
        /*c_mod=*/(short)0, acc, /*reuse_a=*/false, /*reuse_b=*/false);
  }

  // ---- row max (lane-local 64 values + one partner shuffle) ----------------
  float mx = -3.402823466e38f;
#pragma unroll
  for (int kk = 0; kk < 32; ++kk) mx = fmaxf(mx, fmaxf(xr[kk][0], xr[kk][1]));
  mx = fmaxf(mx, __shfl_xor(mx, 16));

  // ---- s1 = sum e^(x-mx), s2 = sum e^(x-mx)*(x-mx) -------------------------
  float s1 = 0.f, s2 = 0.f;
#pragma unroll
  for (int kk = 0; kk < 32; ++kk) {
    float t0 = xr[kk][0] - mx, t1 = xr[kk][1] - mx;
    float e0 = __expf(t0),     e1 = __expf(t1);
    s1 += e0 + e1;
    s2 += e0 * t0 + e1 * t1;
  }
  s1 += __shfl_xor(s1, 16);
  s2 += __shfl_xor(s2, 16);
  const float ls1 = __logf(s1);
  const float H   = s2 / s1 - ls1;          // H_hat = sum p*log p  (<= 0)

  // ---- extract LTS[m] from the D fragment ----------------------------------
  // lane l holds D[r + 8*half][l&15] in acc[r]; all columns of D are equal.
  float t01 = (m & 1) ? acc[1] : acc[0];
  float t23 = (m & 1) ? acc[3] : acc[2];
  float t45 = (m & 1) ? acc[5] : acc[4];
  float t67 = (m & 1) ? acc[7] : acc[6];
  float t03 = (m & 2) ? t23 : t01;
  float t47 = (m & 2) ? t67 : t45;
  float sel = (m & 4) ? t47 : t03;          // = LTS[(m&7) + 8*half]
  float oth = __shfl_xor(sel, 16);
  float lts = (((m >> 3) == half) ? sel : oth);

  // ---- temperature ---------------------------------------------------------
  const float wH = wHp[0], bb = bp[0];
  float a  = lts + wH * (H / 4.8520302639196171f) + bb;   // log(128)
  float sp = fmaxf(a, 0.f) + log1pf(__expf(-fabsf(a)));   // stable softplus
  float T  = fmaxf(sp, 1.1920928955078125e-7f);           // FLT_EPSILON clamp
  float invT = 1.0f / T;

  // ---- scaled logits + second log-softmax ----------------------------------
  float mx2 = mx * invT;
  float s3  = 0.f;
#pragma unroll
  for (int kk = 0; kk < 32; ++kk) {
    float sc0 = xr[kk][0] * invT;
    float sc1 = xr[kk][1] * invT;
    s3 += __expf(sc0 - mx2) + __expf(sc1 - mx2);
    v2f o; o[0] = sc0; o[1] = sc1;
    *(v2f*)(out + rowOff + 4 * kk + cb) = o;
  }
  s3 += __shfl_xor(s3, 16);
  float lse2 = mx2 + __logf(s3);

  // ---- NLL at label, block reduction ---------------------------------------
  int   lab  = labels[row];
  float xlab = X[rowOff + lab];             // row is L0-hot; one extra dword
  float nll  = lse2 - xlab * invT;
  float c    = (half == 0) ? nll : 0.f;     // count each row once
  c = wave_sum32(c);
  if (lane == 0) sred[wave] = c;
  __syncthreads();
  if (wave == 0) {
    float u = (lane < 8) ? sred[lane] : 0.f;
    u += __shfl_xor(u, 4);
    u += __shfl_xor(u, 2);
    u += __shfl_xor(u, 1);
    if (lane == 0) blockSums[blockIdx.x] = u;
  }
}

__global__ __launch_bounds__(256) void ats_finish(
    const float* __restrict__ blockSums, float* __restrict__ out)
{
  __shared__ float sr[8];
  const int tid = threadIdx.x, lane = tid & 31, wave = tid >> 5;
  float v = 0.f;
  for (int i = tid; i < NBLOCKS; i += 256) v += blockSums[i];
  v = wave_sum32(v);
  if (lane == 0) sr[wave] = v;
  __syncthreads();
  if (wave == 0) {
    float u = (lane < 8) ? sr[lane] : 0.f;
    u += __shfl_xor(u, 4);
    u += __shfl_xor(u, 2);
    u += __shfl_xor(u, 1);
    if (lane == 0) out[(size_t)NROWS * NCOLS] = u * (1.0f / NROWS);
  }
}

extern "C" void kernel_launch(void* const* d_in, const int* in_sizes, int n_in,
                              void* d_out, int out_size, void* d_ws, size_t ws_size,
                              hipStream_t stream) {
  const float* X      = (const float*)d_in[0];
  const int*   labels = (const int*)  d_in[1];
  const float* wL     = (const float*)d_in[2];
  const float* wH     = (const float*)d_in[3];
  const float* b      = (const float*)d_in[4];
  float* out = (float*)d_out;
  float* bs  = (float*)d_ws;                 // 4096 floats of scratch

  ats_main  <<<NBLOCKS, 256, 0, stream>>>(X, labels, wL, wH, b, out, bs);
  ats_finish<<<1,       256, 0, stream>>>(bs, out);
}